// LowRankLinearKernel_89859305767561
// MI455X (gfx1250) — compile-verified
//
#include <hip/hip_runtime.h>

#define HH 256
#define WW 256
#define BATCH 16
#define CIN 64
#define COUT 64
#define RANK 64

typedef float v2f __attribute__((ext_vector_type(2)));
typedef float v4f __attribute__((ext_vector_type(4)));
typedef float v8f __attribute__((ext_vector_type(8)));

// ---------------------------------------------------------------------------
// Kernel 1: per-(b,i) plane moments  My = sum v*y_h , Mx = sum v*x_w
// One block per plane (1024 blocks), streaming 128-bit NT loads (256 MB read).
// ---------------------------------------------------------------------------
__global__ __launch_bounds__(256) void moments_kernel(
    const float* __restrict__ v, float* __restrict__ my, float* __restrict__ mx) {
  const int p = blockIdx.x;                       // plane index b*CIN + i
  const v4f* vp = (const v4f*)(v + (size_t)p * (HH * WW));
  const float dd = 2.0f / 255.0f;
  const int t = threadIdx.x;
  float sy = 0.0f, sx = 0.0f;
#pragma unroll 4
  for (int it = 0; it < 64; ++it) {
    const int f = it * 256 + t;                   // float4 index in plane
    const int h = f >> 6;                         // 64 float4 per row
    const int w0 = (f & 63) << 2;
    const float y = -1.0f + dd * (float)h;
    const float x0 = -1.0f + dd * (float)w0;
    const v4f val = __builtin_nontemporal_load(&vp[f]);
    sy += y * (val.x + val.y + val.z + val.w);
    sx += val.x * x0 + val.y * (x0 + dd) + val.z * (x0 + 2.0f * dd) +
          val.w * (x0 + 3.0f * dd);
  }
  // wave32 reduction
#pragma unroll
  for (int off = 16; off > 0; off >>= 1) {
    sy += __shfl_down(sy, off, 32);
    sx += __shfl_down(sx, off, 32);
  }
  __shared__ float red[16];
  const int lane = t & 31, wv = t >> 5;
  if (lane == 0) { red[2 * wv] = sy; red[2 * wv + 1] = sx; }
  __syncthreads();
  if (t == 0) {
    float ty = 0.0f, tx = 0.0f;
#pragma unroll
    for (int k = 0; k < 8; ++k) { ty += red[2 * k]; tx += red[2 * k + 1]; }
    my[p] = ty;
    mx[p] = tx;
  }
}

// ---------------------------------------------------------------------------
// Kernel 2: tiny GEMM chain via V_WMMA_F32_16X16X4_F32 (exact fp32).
//   inner[16,64] = Mcat[16,128] x Wmat[128,64]        (Wmat from psi * dx)
//   A[16,64]     = inner x PhiY[64,64],  B = inner x PhiX[64,64]
// 1 block, 4 waves; wave wv owns output columns [16*wv, 16*wv+16).
// A-fragment layout (32-bit, 16x4): lane m=lane&15, K = k0 + 2*(lane>>4) + j.
// B/C/D: lane n=lane&15 selects column; row = j + 8*(lane>>4) for C/D.
// ---------------------------------------------------------------------------
__global__ __launch_bounds__(128) void smallgemm_kernel(
    const float* __restrict__ my, const float* __restrict__ mx,
    const float* __restrict__ psi, const float* __restrict__ phi,
    float* __restrict__ Aout, float* __restrict__ Bout) {
  __shared__ float mcat[BATCH * 2 * CIN];         // [b][128]  (My | Mx)
  __shared__ float inner_lds[BATCH * RANK];       // [b][r]
  const int tid = threadIdx.x;

  for (int idx = tid; idx < BATCH * 128; idx += 128) {
    const int bb = idx >> 7, ii = idx & 127;
    mcat[idx] = (ii < CIN) ? my[bb * CIN + ii] : mx[bb * CIN + (ii - CIN)];
  }
  __syncthreads();

  const int lane = tid & 31;
  const int wv = tid >> 5;                        // N-tile index
  const int n = lane & 15;                        // column / M-row within half
  const int g = lane >> 4;                        // lane half
  const float dx = 2.0f / 255.0f;

  // ---- Phase A: inner tile, K = 128 (32 wmma steps) ----
  v8f acc = {};
#pragma unroll
  for (int k0 = 0; k0 < 2 * CIN; k0 += 4) {
    const int ka = k0 + 2 * g;
    v2f a, bf;
    a.x = mcat[n * 128 + ka];
    a.y = mcat[n * 128 + ka + 1];
    const int r = wv * 16 + n;
    // Wmat[k][r] = dx * psi[r][k & 63][k >> 6]   (psi row stride = 128)
    bf.x = dx * psi[r * 128 + ((ka & 63) << 1) + (ka >> 6)];
    const int kb = ka + 1;
    bf.y = dx * psi[r * 128 + ((kb & 63) << 1) + (kb >> 6)];
    acc = __builtin_amdgcn_wmma_f32_16x16x4_f32(
        false, a, false, bf, (short)0, acc, false, false);
  }

  // ---- Phase B: spill inner tile to LDS ----
#pragma unroll
  for (int j = 0; j < 8; ++j)
    inner_lds[(j + 8 * g) * RANK + wv * 16 + n] = acc[j];
  __syncthreads();

  // ---- Phase C: A = inner x PhiY, B = inner x PhiX (K = 64, shared A-frag) ----
  v8f cA = {}, cB = {};
#pragma unroll
  for (int k0 = 0; k0 < RANK; k0 += 4) {
    const int ka = k0 + 2 * g;
    v2f a, bY, bX;
    a.x = inner_lds[n * RANK + ka];
    a.y = inner_lds[n * RANK + ka + 1];
    const int o = wv * 16 + n;
    // PhiY[r][o] = phi[o][r][0], PhiX[r][o] = phi[o][r][1]; phi row stride 128
    bY.x = phi[o * 128 + ka * 2 + 0];
    bY.y = phi[o * 128 + (ka + 1) * 2 + 0];
    bX.x = phi[o * 128 + ka * 2 + 1];
    bX.y = phi[o * 128 + (ka + 1) * 2 + 1];
    cA = __builtin_amdgcn_wmma_f32_16x16x4_f32(
        false, a, false, bY, (short)0, cA, false, false);
    cB = __builtin_amdgcn_wmma_f32_16x16x4_f32(
        false, a, false, bX, (short)0, cB, false, false);
  }

  // ---- Phase D: write A[b][o], B[b][o] ----
#pragma unroll
  for (int j = 0; j < 8; ++j) {
    const int m = j + 8 * g;
    const int o = wv * 16 + n;
    Aout[m * COUT + o] = cA[j];
    Bout[m * COUT + o] = cB[j];
  }
}

// ---------------------------------------------------------------------------
// Kernel 3: u[b,o,h,w] = A[b,o]*y_h + B[b,o]*x_w  (256 MB NT stores)
// One block per (b,o) plane.
// ---------------------------------------------------------------------------
__global__ __launch_bounds__(256) void expand_kernel(
    const float* __restrict__ Aarr, const float* __restrict__ Barr,
    float* __restrict__ u) {
  const int p = blockIdx.x;                       // b*COUT + o
  const float Ao = Aarr[p];
  const float Bo = Barr[p];
  v4f* up = (v4f*)(u + (size_t)p * (HH * WW));
  const float dd = 2.0f / 255.0f;
  const int t = threadIdx.x;
#pragma unroll 4
  for (int it = 0; it < 64; ++it) {
    const int f = it * 256 + t;
    const int h = f >> 6;
    const int w0 = (f & 63) << 2;
    const float y = -1.0f + dd * (float)h;
    const float x0 = -1.0f + dd * (float)w0;
    const float base = Ao * y;
    v4f o4;
    o4.x = base + Bo * x0;
    o4.y = base + Bo * (x0 + dd);
    o4.z = base + Bo * (x0 + 2.0f * dd);
    o4.w = base + Bo * (x0 + 3.0f * dd);
    __builtin_nontemporal_store(o4, &up[f]);
  }
}

// ---------------------------------------------------------------------------
extern "C" void kernel_launch(void* const* d_in, const int* in_sizes, int n_in,
                              void* d_out, int out_size, void* d_ws, size_t ws_size,
                              hipStream_t stream) {
  const float* v   = (const float*)d_in[0];   // [16,64,256,256]
  const float* psi = (const float*)d_in[1];   // [64,64,2]
  const float* phi = (const float*)d_in[2];   // [64,64,2]
  float* out = (float*)d_out;                 // [16,64,256,256]
  float* ws  = (float*)d_ws;

  float* my   = ws;                           // 1024 floats
  float* mx   = ws + 1024;                    // 1024 floats
  float* Aarr = ws + 2048;                    // 1024 floats
  float* Barr = ws + 3072;                    // 1024 floats

  moments_kernel<<<BATCH * CIN, 256, 0, stream>>>(v, my, mx);
  smallgemm_kernel<<<1, 128, 0, stream>>>(my, mx, psi, phi, Aarr, Barr);
  expand_kernel<<<BATCH * COUT, 256, 0, stream>>>(Aarr, Barr, out);
}